// DeviceGNN_28604482191526
// MI455X (gfx1250) — compile-verified
//
#include <hip/hip_runtime.h>

typedef float v2f __attribute__((ext_vector_type(2)));
typedef float v8f __attribute__((ext_vector_type(8)));

#define NN 50000
#define NE 800000
#define D 96

__global__ void zero_f32(float* __restrict__ p, int n) {
  int t = blockIdx.x * blockDim.x + threadIdx.x;
  if (t < n) p[t] = 0.0f;
}

__global__ void count_indeg(const int* __restrict__ edst, float* __restrict__ indeg, int nE) {
  int t = blockIdx.x * blockDim.x + threadIdx.x;
  if (t < nE) atomicAdd(&indeg[edst[t]], 1.0f);
}

__global__ void indeg_to_inv(float* __restrict__ indeg, int n) {
  int t = blockIdx.x * blockDim.x + threadIdx.x;
  if (t < n) indeg[t] = 1.0f / fmaxf(indeg[t], 1.0f);
}

__global__ void gather_emb(const int* __restrict__ degree, const float* __restrict__ emb,
                           float* __restrict__ feat, int total) {
  int t = blockIdx.x * blockDim.x + threadIdx.x;
  if (t >= total) return;
  int n = t / D;
  int d = t - n * D;
  feat[t] = emb[degree[n] * D + d];
}

// one wave32 per edge: 32 lanes x 3 dims = 96 floats, coalesced loads + f32 atomics
__global__ void edge_scatter(const int* __restrict__ esrc, const int* __restrict__ edst,
                             const float* __restrict__ feat, float* __restrict__ agg, int nE) {
  int wid = (blockIdx.x * blockDim.x + threadIdx.x) >> 5;
  int lane = threadIdx.x & 31;
  if (wid >= nE) return;
  int s = esrc[wid];
  int d = edst[wid];
  const float* fs = feat + (size_t)s * D;
  float* ad = agg + (size_t)d * D;
#pragma unroll
  for (int j = 0; j < 3; ++j) {
    float v = fs[lane + 32 * j];
    atomicAdd(&ad[lane + 32 * j], v);
  }
}

// one wave per 16-row tile: Y[16 x 96] = X @ Ws + (agg*inv) @ Wn + b via fp32 WMMA
__global__ void sage_gemm_wmma(const float* __restrict__ X, const float* __restrict__ A,
                               const float* __restrict__ inv, const float* __restrict__ Ws,
                               const float* __restrict__ Wn, const float* __restrict__ bias,
                               float* __restrict__ Y) {
  const int lane = threadIdx.x & 31;
  const int wave = (blockIdx.x * blockDim.x + threadIdx.x) >> 5;
  const int nRowTiles = NN / 16;  // 3125, exact
  if (wave >= nRowTiles) return;

  const int row0 = wave * 16;
  const int m  = lane & 15;         // M (A) / N (B,C) index within tile
  const int kk = (lane >> 4) << 1;  // K sub-offset: 0 for lanes 0-15, 2 for 16-31

  const float s = inv[row0 + m];
  const float* xrow = X + (size_t)(row0 + m) * D;
  const float* arow = A + (size_t)(row0 + m) * D;

  v8f acc[6] = {};  // 6 column tiles of 16, accumulated in fp32

  for (int k0 = 0; k0 < D; k0 += 4) {
    v2f a_self, a_agg;
    a_self.x = xrow[k0 + kk];
    a_self.y = xrow[k0 + kk + 1];
    a_agg.x  = arow[k0 + kk] * s;
    a_agg.y  = arow[k0 + kk + 1] * s;

    const float* wsk = Ws + (size_t)(k0 + kk) * D + m;  // W[k][n], row-major
    const float* wnk = Wn + (size_t)(k0 + kk) * D + m;
#pragma unroll
    for (int c = 0; c < 6; ++c) {
      v2f bs, bn;
      bs.x = wsk[c * 16];
      bs.y = wsk[c * 16 + D];
      bn.x = wnk[c * 16];
      bn.y = wnk[c * 16 + D];
      acc[c] = __builtin_amdgcn_wmma_f32_16x16x4_f32(false, a_self, false, bs,
                                                     (short)0, acc[c], false, false);
      acc[c] = __builtin_amdgcn_wmma_f32_16x16x4_f32(false, a_agg, false, bn,
                                                     (short)0, acc[c], false, false);
    }
  }

  const int mrow = (lane >> 4) * 8;  // C/D layout: VGPR r -> M = r (+8 for upper lanes)
#pragma unroll
  for (int c = 0; c < 6; ++c) {
    float bv = bias[c * 16 + m];
#pragma unroll
    for (int r = 0; r < 8; ++r) {
      Y[(size_t)(row0 + mrow + r) * D + c * 16 + m] = acc[c][r] + bv;
    }
  }
}

extern "C" void kernel_launch(void* const* d_in, const int* in_sizes, int n_in,
                              void* d_out, int out_size, void* d_ws, size_t ws_size,
                              hipStream_t stream) {
  const int* degree = (const int*)d_in[0];
  const int* esrc   = (const int*)d_in[1];
  const int* edst   = (const int*)d_in[2];
  const float* emb  = (const float*)d_in[3];
  const float* Ws[3] = {(const float*)d_in[4], (const float*)d_in[7],  (const float*)d_in[10]};
  const float* Wn[3] = {(const float*)d_in[5], (const float*)d_in[8],  (const float*)d_in[11]};
  const float* bb[3] = {(const float*)d_in[6], (const float*)d_in[9],  (const float*)d_in[12]};
  float* out = (float*)d_out;

  const size_t featElems = (size_t)NN * D;  // 4.8M floats
  float* featA = (float*)d_ws;
  float* featB = featA + featElems;
  float* agg   = featB + featElems;
  float* inv   = agg + featElems;

  // in-degree -> inverse (clamped to 1)
  zero_f32<<<(NN + 255) / 256, 256, 0, stream>>>(inv, NN);
  count_indeg<<<(NE + 255) / 256, 256, 0, stream>>>(edst, inv, NE);
  indeg_to_inv<<<(NN + 255) / 256, 256, 0, stream>>>(inv, NN);

  // feat = emb[degree]
  gather_emb<<<((int)featElems + 255) / 256, 256, 0, stream>>>(degree, emb, featA, (int)featElems);

  float* fin  = featA;
  float* fout = featB;
  const int gemmThreads = (NN / 16) * 32;  // one wave per row tile
  for (int l = 0; l < 3; ++l) {
    zero_f32<<<((int)featElems + 255) / 256, 256, 0, stream>>>(agg, (int)featElems);
    edge_scatter<<<(NE * 32 + 255) / 256, 256, 0, stream>>>(esrc, edst, fin, agg, NE);
    float* dst = (l == 2) ? out : fout;
    sage_gemm_wmma<<<(gemmThreads + 255) / 256, 256, 0, stream>>>(fin, agg, inv,
                                                                  Ws[l], Wn[l], bb[l], dst);
    float* t = fin; fin = fout; fout = t;
  }
}